// BilinearFullSymLoss_78589311582542
// MI455X (gfx1250) — compile-verified
//
#include <hip/hip_runtime.h>
#include <hip/hip_bf16.h>

// Problem constants (match reference setup_inputs: B=32, h=w=512, C=10)
#define HH 512
#define WW 512
#define NB 32
#define CC 10.0f
#define BLOCKS_PER_SAMPLE 64
#define ROWS_PER_BLOCK (HH / BLOCKS_PER_SAMPLE)      // 8
#define NPART (NB * BLOCKS_PER_SAMPLE)               // 2048

typedef float v2f __attribute__((ext_vector_type(2)));
typedef float v8f __attribute__((ext_vector_type(8)));

// ---------------------------------------------------------------------------
// Kernel 1: per-(sample, row-band) masked sum of squared deltas.
// 4 pixels per thread: one b128 base load + register-reused stencil taps.
// Tap addresses are clamped (identity for in-mask pixels) so all loads are
// unconditional and in-bounds; masked-out lanes discard via select.
// ---------------------------------------------------------------------------
__global__ __launch_bounds__(256) void bls_partial(const float* __restrict__ grid,
                                                   const float* __restrict__ gt,
                                                   const float* __restrict__ gd,
                                                   float* __restrict__ partial) {
  const int b   = blockIdx.y;        // sample
  const int blk = blockIdx.x;        // row band
  const int t   = threadIdx.x;

  // Per-sample scalars (uniform across block -> SALU-friendly)
  const float dx = -CC * gt[2 * b + 0];
  const float dy =  CC * gt[2 * b + 1];
  const float sx = gd[2 * b + 0];
  const float sy = gd[2 * b + 1];

  const float dx1f = floorf(dx), dy1f = floorf(dy);
  const int   dx1 = (int)dx1f,   dy1 = (int)dy1f;
  const float fx = dx - dx1f,    fy = dy - dy1f;
  const float w22 = fx * fy;
  const float w21 = fx * (1.0f - fy);
  const float w12 = (1.0f - fx) * fy;
  const float w11 = (1.0f - fx) * (1.0f - fy);
  const bool  pos = dx > 0.0f;
  const int rows = HH - dy1 - 1;
  const int cols = pos ? (WW - dx1 - 1) : (WW + dx1);

  const float* __restrict__ g0 = grid + (size_t)b * 2u * HH * WW;
  const float* __restrict__ g1 = g0 + (size_t)HH * WW;

  const int r0   = blk * ROWS_PER_BLOCK;
  const int jg   = (t & 127) << 2;   // 4-pixel column group base (16B aligned)
  const int rsub = t >> 7;           // 0/1: two rows in flight per iteration

  float acc = 0.0f;

  #pragma unroll
  for (int k = 0; k < 4; ++k) {
    const int i = r0 + 2 * k + rsub;

    // gfx1250 stream prefetch (global_prefetch_b8) of the next base rows
    if (k < 3) {
      __builtin_prefetch(g0 + (size_t)(i + 2) * WW + jg, 0, 1);
      __builtin_prefetch(g1 + (size_t)(i + 2) * WW + jg, 0, 1);
    }

    if (pos) {
      // delta = g[i,j] - bilinear sample at (i+dy, j+dx)
      const int iA = min(i + dy1, HH - 1);       // identity when i < rows
      const int iB = min(i + dy1 + 1, HH - 1);
      const float4 b0 = *(const float4*)(g0 + (size_t)i * WW + jg);
      const float4 b1 = *(const float4*)(g1 + (size_t)i * WW + jg);
      float tA0[5], tB0[5], tA1[5], tB1[5];
      #pragma unroll
      for (int m = 0; m < 5; ++m) {
        const int tc = min(jg + dx1 + m, WW - 1);  // identity for in-mask pixels
        tA0[m] = g0[(size_t)iA * WW + tc];
        tB0[m] = g0[(size_t)iB * WW + tc];
        tA1[m] = g1[(size_t)iA * WW + tc];
        tB1[m] = g1[(size_t)iB * WW + tc];
      }
      const float base0[4] = {b0.x, b0.y, b0.z, b0.w};
      const float base1[4] = {b1.x, b1.y, b1.z, b1.w};
      #pragma unroll
      for (int p = 0; p < 4; ++p) {
        const float d0 = base0[p] - (w11 * tA0[p] + w21 * tA0[p + 1] +
                                     w12 * tB0[p] + w22 * tB0[p + 1]);
        const float d1 = base1[p] - (w11 * tA1[p] + w21 * tA1[p + 1] +
                                     w12 * tB1[p] + w22 * tB1[p + 1]);
        const float v = d0 * sy + d1 * sx;
        const bool  ok = (i < rows) && (jg + p < cols);
        acc += ok ? v * v : 0.0f;
      }
    } else {
      // d_neg = fx*a2 + (1-fx)*a1 - fy*s_top - (1-fy)*s_bot (weights telescoped)
      const int iB = min(i + dy1 + 1, HH - 1);
      const int iP = min(i + 1, HH - 1);
      // a-taps: row iB, columns jg..jg+4 -> aligned b128 + 1 clamped scalar
      const float4 a40 = *(const float4*)(g0 + (size_t)iB * WW + jg);
      const float4 a41 = *(const float4*)(g1 + (size_t)iB * WW + jg);
      const int   jc4  = min(jg + 4, WW - 1);
      const float aB0[5] = {a40.x, a40.y, a40.z, a40.w, g0[(size_t)iB * WW + jc4]};
      const float aB1[5] = {a41.x, a41.y, a41.z, a41.w, g1[(size_t)iB * WW + jc4]};
      // s-taps: rows i / i+1, columns jg - dx1 .. +3 (dx1 < 0 -> shifted right)
      float sT0[4], sB0[4], sT1[4], sB1[4];
      #pragma unroll
      for (int m = 0; m < 4; ++m) {
        const int tc = min(jg - dx1 + m, WW - 1);  // identity for in-mask pixels
        sT0[m] = g0[(size_t)i  * WW + tc];
        sB0[m] = g0[(size_t)iP * WW + tc];
        sT1[m] = g1[(size_t)i  * WW + tc];
        sB1[m] = g1[(size_t)iP * WW + tc];
      }
      #pragma unroll
      for (int p = 0; p < 4; ++p) {
        const float d0 = fx * aB0[p + 1] + (1.0f - fx) * aB0[p]
                       - fy * sT0[p]     - (1.0f - fy) * sB0[p];
        const float d1 = fx * aB1[p + 1] + (1.0f - fx) * aB1[p]
                       - fy * sT1[p]     - (1.0f - fy) * sB1[p];
        const float v = d0 * sy + d1 * sx;
        const bool  ok = (i < rows) && (jg + p < cols);
        acc += ok ? v * v : 0.0f;
      }
    }
  }

  // Deterministic block reduction: wave32 shuffle tree + LDS across 8 waves.
  for (int off = 16; off > 0; off >>= 1)
    acc += __shfl_xor(acc, off, 32);

  __shared__ float red[8];
  const int wave = t >> 5, lane = t & 31;
  if (lane == 0) red[wave] = acc;
  __syncthreads();
  if (t == 0) {
    float s = 0.0f;
    #pragma unroll
    for (int wv = 0; wv < 8; ++wv) s += red[wv];
    partial[b * BLOCKS_PER_SAMPLE + blk] = s;
  }
}

// ---------------------------------------------------------------------------
// Kernel 2: deterministic final reduction on the matrix pipe.
// V_WMMA_F32_16X16X4_F32 with A = ones(16x4): D[m,n] = sum_k B[k,n] + C[m,n],
// i.e. each WMMA folds 64 scaled partials into a row-replicated f32
// accumulator (exact f32 adds, fixed order). One wave, 32 iterations.
// ---------------------------------------------------------------------------
__device__ __forceinline__ float bls_scale(const float* __restrict__ gt, int s) {
  const float dx = -CC * gt[2 * s + 0];
  const float dy =  CC * gt[2 * s + 1];
  const int dx1 = (int)floorf(dx);
  const int dy1 = (int)floorf(dy);
  const int rows = HH - dy1 - 1;
  const int cols = (dx > 0.0f) ? (WW - dx1 - 1) : (WW + dx1);
  return 1.0f / ((float)(rows * cols) * (float)NB);
}

__global__ __launch_bounds__(32) void bls_finalize(const float* __restrict__ partial,
                                                   const float* __restrict__ gt,
                                                   float* __restrict__ out) {
  const int lane = threadIdx.x;          // one full wave32, EXEC all ones

  v2f a; a.x = 1.0f; a.y = 1.0f;         // A = ones(16x4)
  v8f c = {};                            // C/D accumulator (16x16 f32)

  // 64 partials per iteration; sample index (= it) is uniform per iteration.
  for (int it = 0; it < NPART / 64; ++it) {
    const int i0 = it * 64 + lane;
    const int i1 = i0 + 32;
    const float sc = bls_scale(gt, it);  // both halves belong to sample `it`
    v2f bm;
    bm.x = partial[i0] * sc;
    bm.y = partial[i1] * sc;
    // (neg_a, A, neg_b, B, c_mod, C, reuse_a, reuse_b)
    c = __builtin_amdgcn_wmma_f32_16x16x4_f32(false, a, false, bm,
                                              (short)0, c, false, false);
  }

  // Every row of D is identical; VGPR0 holds rows M=0 (lanes 0-15) and M=8
  // (lanes 16-31) -> lane-sum of c[0] == 2 * total.
  float v = c[0];
  for (int off = 16; off > 0; off >>= 1)
    v += __shfl_xor(v, off, 32);
  if (lane == 0) out[0] = v * 0.5f;
}

// ---------------------------------------------------------------------------
extern "C" void kernel_launch(void* const* d_in, const int* in_sizes, int n_in,
                              void* d_out, int out_size, void* d_ws, size_t ws_size,
                              hipStream_t stream) {
  (void)in_sizes; (void)n_in; (void)out_size; (void)ws_size;
  const float* grid = (const float*)d_in[0];   // [32, 2, 512, 512] f32
  const float* gt   = (const float*)d_in[1];   // [32, 2] f32
  const float* gd   = (const float*)d_in[2];   // [32, 2] f32
  float* out        = (float*)d_out;           // [1] f32
  float* partial    = (float*)d_ws;            // NPART floats (8 KiB)

  dim3 g(BLOCKS_PER_SAMPLE, NB);
  bls_partial<<<g, 256, 0, stream>>>(grid, gt, gd, partial);
  bls_finalize<<<1, 32, 0, stream>>>(partial, gt, out);
}